// GCNModel_11914239279899
// MI455X (gfx1250) — compile-verified
//
#include <hip/hip_runtime.h>

#define NNODES 50000
#define NEDGES 800000
#define DIM 128
#define NLAYERS 4

typedef __attribute__((ext_vector_type(2))) float v2f;
typedef __attribute__((ext_vector_type(8))) float v8f;

// ---------------------------------------------------------------------------
// Degree accumulation: one thread per edge, float atomics on deg arrays.
// ---------------------------------------------------------------------------
__global__ __launch_bounds__(256) void deg_kernel(const int* __restrict__ src,
                                                  const int* __restrict__ dst,
                                                  float* __restrict__ outdeg,
                                                  float* __restrict__ indeg, int E) {
  int e = blockIdx.x * blockDim.x + threadIdx.x;
  if (e < E) {
    unsafeAtomicAdd(&outdeg[src[e]], 1.0f);
    unsafeAtomicAdd(&indeg[dst[e]], 1.0f);
  }
}

// deg -> rsqrt(max(deg,1)) in place
__global__ __launch_bounds__(256) void norm_kernel(float* __restrict__ outdeg,
                                                   float* __restrict__ indeg, int n) {
  int i = blockIdx.x * blockDim.x + threadIdx.x;
  if (i < n) {
    outdeg[i] = rsqrtf(fmaxf(outdeg[i], 1.0f));
    indeg[i]  = rsqrtf(fmaxf(indeg[i], 1.0f));
  }
}

// ---------------------------------------------------------------------------
// Edge scatter: one wave32 per edge. Lane l handles 4 consecutive floats
// (coalesced 512B float4 gather of h[src], scaled by norm_out[src], native
// f32 atomic adds into agg[dst]).  This is the HBM-bound hot loop.
// ---------------------------------------------------------------------------
__global__ __launch_bounds__(256) void scatter_kernel(const float* __restrict__ h,
                                                      const float* __restrict__ norm_out,
                                                      const int* __restrict__ src,
                                                      const int* __restrict__ dst,
                                                      float* __restrict__ agg, int E) {
  int wave = (int)((blockIdx.x * (unsigned)blockDim.x + threadIdx.x) >> 5);
  int lane = threadIdx.x & 31;
  if (wave >= E) return;
  int s = src[wave];
  int d = dst[wave];
  float ns = norm_out[s];
  float4 v = ((const float4*)(h + (size_t)s * DIM))[lane];
  float* ap = agg + (size_t)d * DIM + lane * 4;
  unsafeAtomicAdd(ap + 0, v.x * ns);
  unsafeAtomicAdd(ap + 1, v.y * ns);
  unsafeAtomicAdd(ap + 2, v.z * ns);
  unsafeAtomicAdd(ap + 3, v.w * ns);
}

// ---------------------------------------------------------------------------
// Fused per-layer: g = (agg * norm_in) @ W + b ; LayerNorm ; ReLU ; +h
// Block = 16 node rows x 256 threads (8 waves). Wave w computes the 16x16
// output tile at columns [16w,16w+16) via 32x V_WMMA_F32_16X16X4_F32.
// A is staged in LDS with stride DIM+4 (bank-conflict-free, 16B aligned rows).
// ---------------------------------------------------------------------------
#define ASTRIDE (DIM + 4)

__global__ __launch_bounds__(256) void gcn_layer_kernel(
    const float* __restrict__ agg, const float* __restrict__ norm_in,
    const float* __restrict__ h_in, const float* __restrict__ W,
    const float* __restrict__ bias, const float* __restrict__ gamma,
    const float* __restrict__ beta, float* __restrict__ h_out, int n_nodes) {
  __shared__ float Asm[16 * ASTRIDE];  // scaled aggregate rows (A operand)
  __shared__ float Gsm[16 * ASTRIDE];  // GEMM result tile for LayerNorm

  const int row0 = blockIdx.x * 16;
  const int tid  = threadIdx.x;
  const int wave = tid >> 5;   // 0..7
  const int lane = tid & 31;

  // ---- stage A = agg * norm_in into LDS (2048 floats, 2 float4 per thread)
  for (int i = tid; i < 16 * (DIM / 4); i += 256) {
    int r  = i >> 5;       // row within tile
    int c4 = i & 31;       // float4 column
    int gr = row0 + r;
    float4 v = make_float4(0.f, 0.f, 0.f, 0.f);
    float sc = 0.f;
    if (gr < n_nodes) {
      v  = ((const float4*)(agg + (size_t)gr * DIM))[c4];
      sc = norm_in[gr];
    }
    float* ar = &Asm[r * ASTRIDE + c4 * 4];
    ar[0] = v.x * sc;
    ar[1] = v.y * sc;
    ar[2] = v.z * sc;
    ar[3] = v.w * sc;
  }
  __syncthreads();

  // ---- fp32 WMMA: D(16x16) += A(16x4) x B(4x16), K = 128 -> 32 steps
  const int n0   = wave * 16;      // this wave's output column tile
  const int half = lane >> 4;      // 0: K pair {0,1}, 1: K pair {2,3}
  const int ml   = lane & 15;      // M index (A) / N index (B)
  v8f acc = {};
  for (int t = 0; t < DIM / 4; ++t) {
    int k0 = t * 4 + half * 2;
    v2f a, b;
    a.x = Asm[ml * ASTRIDE + k0];
    a.y = Asm[ml * ASTRIDE + k0 + 1];
    b.x = W[(size_t)k0 * DIM + n0 + ml];
    b.y = W[(size_t)(k0 + 1) * DIM + n0 + ml];
    acc = __builtin_amdgcn_wmma_f32_16x16x4_f32(false, a, false, b,
                                                (short)0, acc, false, false);
  }

  // ---- epilogue part 1: +bias, park tile in LDS for row-wise LayerNorm
  for (int v = 0; v < 8; ++v) {
    int r = v + half * 8;        // row of this accumulator element
    int c = n0 + ml;             // column
    Gsm[r * ASTRIDE + c] = acc[v] + bias[c];
  }
  __syncthreads();

  // ---- epilogue part 2: LayerNorm + ReLU + residual. Wave w owns rows 2w,2w+1.
  for (int rr = 0; rr < 2; ++rr) {
    int r  = wave * 2 + rr;
    int gr = row0 + r;
    if (gr >= n_nodes) continue;
    float x0 = Gsm[r * ASTRIDE + lane];
    float x1 = Gsm[r * ASTRIDE + lane + 32];
    float x2 = Gsm[r * ASTRIDE + lane + 64];
    float x3 = Gsm[r * ASTRIDE + lane + 96];
    float s  = x0 + x1 + x2 + x3;
    float sq = x0 * x0 + x1 * x1 + x2 * x2 + x3 * x3;
    #pragma unroll
    for (int off = 16; off > 0; off >>= 1) {
      s  += __shfl_xor(s, off, 32);
      sq += __shfl_xor(sq, off, 32);
    }
    float mu  = s * (1.0f / DIM);
    float var = sq * (1.0f / DIM) - mu * mu;
    float inv = rsqrtf(var + 1e-5f);
    const float* hrow = h_in + (size_t)gr * DIM;
    float* orow       = h_out + (size_t)gr * DIM;
    float xs[4] = {x0, x1, x2, x3};
    #pragma unroll
    for (int j = 0; j < 4; ++j) {
      int c  = lane + j * 32;
      float g = (xs[j] - mu) * inv * gamma[c] + beta[c];
      g = fmaxf(g, 0.0f);
      orow[c] = g + hrow[c];
    }
  }
}

// ---------------------------------------------------------------------------
extern "C" void kernel_launch(void* const* d_in, const int* in_sizes, int n_in,
                              void* d_out, int out_size, void* d_ws, size_t ws_size,
                              hipStream_t stream) {
  (void)in_sizes; (void)n_in; (void)out_size; (void)ws_size;
  const float* x      = (const float*)d_in[0];
  const int*   src    = (const int*)d_in[1];
  const int*   dst    = (const int*)d_in[2];
  const float* Ws     = (const float*)d_in[3];  // [4][128][128]
  const float* bs     = (const float*)d_in[4];  // [4][128]
  const float* gammas = (const float*)d_in[5];  // [4][128]
  const float* betas  = (const float*)d_in[6];  // [4][128]
  float* out = (float*)d_out;

  // workspace layout: norms (2N) | agg (N*D) | h ping buffer (N*D)  ~51.6 MB
  float* norm_out = (float*)d_ws;
  float* norm_in  = norm_out + NNODES;
  float* agg      = norm_in + NNODES;
  float* h_buf    = agg + (size_t)NNODES * DIM;

  // degrees -> rsqrt norms (recomputed every call; deterministic)
  hipMemsetAsync(norm_out, 0, 2 * (size_t)NNODES * sizeof(float), stream);
  deg_kernel<<<(NEDGES + 255) / 256, 256, 0, stream>>>(src, dst, norm_out, norm_in, NEDGES);
  norm_kernel<<<(NNODES + 255) / 256, 256, 0, stream>>>(norm_out, norm_in, NNODES);

  const float* h = x;
  for (int layer = 0; layer < NLAYERS; ++layer) {
    hipMemsetAsync(agg, 0, (size_t)NNODES * DIM * sizeof(float), stream);
    // one wave per edge -> E*32 threads
    scatter_kernel<<<(int)(((long long)NEDGES * 32 + 255) / 256), 256, 0, stream>>>(
        h, norm_out, src, dst, agg, NEDGES);
    float* hout = (layer == NLAYERS - 1) ? out : h_buf;
    gcn_layer_kernel<<<(NNODES + 15) / 16, 256, 0, stream>>>(
        agg, norm_in, h, Ws + (size_t)layer * DIM * DIM, bs + layer * DIM,
        gammas + layer * DIM, betas + layer * DIM, hout, NNODES);
    h = hout;
  }
}